// Adapter_1949915152418
// MI455X (gfx1250) — compile-verified
//
#include <hip/hip_runtime.h>

#define B_   256
#define L_   197
#define C_   768
#define H_   48
#define NA   10
#define BL   (B_ * L_)          /* 50432 = 394 * 128 */
#define BIAS_ELEMS ((size_t)BL * C_)

typedef __attribute__((ext_vector_type(2))) float v2f;
typedef __attribute__((ext_vector_type(8))) float v8f;

/* gfx1250 async global->LDS copy, 16B per instruction, tracked by ASYNCcnt. */
__device__ __forceinline__ void async_copy_b128(unsigned lds_off, const void* gptr) {
    asm volatile("global_load_async_to_lds_b128 %0, %1, off"
                 :: "v"(lds_off), "v"((unsigned long long)(uintptr_t)gptr)
                 : "memory");
}
__device__ __forceinline__ void wait_async0() {
    asm volatile("s_wait_asynccnt 0x0" ::: "memory");
}

/* ---------------- Kernel 1: mean over L ---------------- */
__global__ void k_mean(const float* __restrict__ X, float* __restrict__ mean_out) {
    int gid = blockIdx.x * 256 + threadIdx.x;          // B_*C_ = 196608 threads exactly
    int b = gid / C_, c = gid % C_;
    const float* p = X + (size_t)b * L_ * C_ + c;
    float s = 0.f;
    for (int l = 0; l < L_; ++l) s += p[(size_t)l * C_];
    mean_out[gid] = s * (1.0f / (float)L_);
}

/* ------- Kernel 2: normalized similarity + per-sample argmax ------- */
__global__ void k_sim(const float* __restrict__ mean_in, const float* __restrict__ key,
                      float* __restrict__ sim, int* __restrict__ psidx) {
    __shared__ float red[256];
    __shared__ float simsh[NA];
    int b = blockIdx.x, t = threadIdx.x;
    float xm0 = mean_in[b * C_ + t];
    float xm1 = mean_in[b * C_ + t + 256];
    float xm2 = mean_in[b * C_ + t + 512];

    red[t] = xm0 * xm0 + xm1 * xm1 + xm2 * xm2;
    __syncthreads();
    for (int s = 128; s > 0; s >>= 1) { if (t < s) red[t] += red[t + s]; __syncthreads(); }
    float rx = rsqrtf(fmaxf(red[0], 1e-12f));
    __syncthreads();

    for (int a = 0; a < NA; ++a) {
        float k0 = key[a * C_ + t], k1 = key[a * C_ + t + 256], k2 = key[a * C_ + t + 512];
        red[t] = xm0 * k0 + xm1 * k1 + xm2 * k2;
        __syncthreads();
        for (int s = 128; s > 0; s >>= 1) { if (t < s) red[t] += red[t + s]; __syncthreads(); }
        float dot = red[0];
        __syncthreads();
        red[t] = k0 * k0 + k1 * k1 + k2 * k2;
        __syncthreads();
        for (int s = 128; s > 0; s >>= 1) { if (t < s) red[t] += red[t + s]; __syncthreads(); }
        float rk = rsqrtf(fmaxf(red[0], 1e-12f));
        __syncthreads();
        if (t == 0) { float sv = dot * rx * rk; sim[b * NA + a] = sv; simsh[a] = sv; }
    }
    __syncthreads();
    if (t == 0) {
        float best = simsh[0]; int bi = 0;
        for (int a = 1; a < NA; ++a) if (simsh[a] > best) { best = simsh[a]; bi = a; }
        psidx[b] = bi;                                  // first-max tie rule
    }
}

/* ------- Kernel 3: majority vote + reduce_sim + idx output ------- */
__global__ void k_vote(const float* __restrict__ sim, const int* __restrict__ psidx,
                       int* __restrict__ major_out, float* __restrict__ out_tail) {
    __shared__ int   cnt[NA];
    __shared__ int   mj;
    __shared__ float red[256];
    int t = threadIdx.x;
    if (t < NA) cnt[t] = 0;
    __syncthreads();
    atomicAdd(&cnt[psidx[t]], 1);
    __syncthreads();
    if (t == 0) {
        int best = cnt[0], bi = 0;
        for (int a = 1; a < NA; ++a) if (cnt[a] > best) { best = cnt[a]; bi = a; }  // ties -> smallest id
        mj = bi; major_out[0] = bi;
    }
    __syncthreads();
    int m = mj;
    red[t] = sim[t * NA + m];
    __syncthreads();
    for (int s = 128; s > 0; s >>= 1) { if (t < s) red[t] += red[t + s]; __syncthreads(); }
    if (t == 0) out_tail[0] = red[0] * (1.0f / (float)B_);   // reduce_sim
    out_tail[1 + t] = (float)m;                              // idx[B,1]
}

/* ------- Kernel 4: fused adapter  out = x + relu(relu(x@W1^T)@W2^T) -------
 * 256 thr = 8 waves; wave owns 16 rows; block owns 128 rows (50432 = 394*128).
 * LDS: w1 [48][768] f32 + w2 [768][48] f32 + h [8][16][48] f32 = 319488 B.
 * Weights staged via gfx1250 async global->LDS (ASYNCcnt), no VGPR transit.
 * f32 WMMA 16x16x4: A frag per lane = float2 at K = 4*ks + 2*(lane>>4)+{0,1},
 * row M = lane&15. B frag symmetric with N = lane&15. D: M = i+8*(lane>>4).
 */
__global__ void k_adapter(const float* __restrict__ X,
                          const float* __restrict__ W1,
                          const float* __restrict__ W2,
                          const int* __restrict__ majorp,
                          float* __restrict__ out) {
    extern __shared__ float smem[];
    float* w1s  = smem;                    // 48*768
    float* w2s  = smem + H_ * C_;          // 768*48
    float* hbuf = smem + 2 * H_ * C_;      // 8*16*48

    int t = threadIdx.x;
    int major = majorp[0];
    const float* w1g = W1 + (size_t)major * H_ * C_;
    const float* w2g = W2 + (size_t)major * C_ * H_;

    /* async-stage both weight matrices into LDS (16B granules) */
    unsigned w1off = (unsigned)(uintptr_t)w1s;
    unsigned w2off = (unsigned)(uintptr_t)w2s;
    for (int i = t; i < (H_ * C_) / 4; i += 256) {
        async_copy_b128(w1off + 16u * (unsigned)i, (const char*)w1g + 16 * (size_t)i);
        async_copy_b128(w2off + 16u * (unsigned)i, (const char*)w2g + 16 * (size_t)i);
    }
    wait_async0();
    __syncthreads();

    int wave = t >> 5, lane = t & 31;
    int rm = lane & 15, half = lane >> 4;
    int row0 = blockIdx.x * 128 + wave * 16;
    float* hb = hbuf + wave * (16 * H_);

    /* ---- GEMM1: h = relu(X[16x768] @ w1^T[768x48]) ---- */
    v8f acc1[3];
    #pragma unroll
    for (int nt = 0; nt < 3; ++nt)
        #pragma unroll
        for (int i = 0; i < 8; ++i) acc1[nt][i] = 0.f;

    const float* xrow = X + (size_t)(row0 + rm) * C_;
    for (int ks = 0; ks < C_ / 4; ++ks) {
        int k = ks * 4 + 2 * half;
        v2f a = *(const v2f*)(xrow + k);
        #pragma unroll
        for (int nt = 0; nt < 3; ++nt) {
            v2f bf = *(const v2f*)(w1s + (nt * 16 + rm) * C_ + k);
            acc1[nt] = __builtin_amdgcn_wmma_f32_16x16x4_f32(
                false, a, false, bf, (short)0, acc1[nt], false, false);
        }
    }

    /* relu + transpose h through wave-private LDS */
    #pragma unroll
    for (int nt = 0; nt < 3; ++nt)
        #pragma unroll
        for (int i = 0; i < 8; ++i) {
            int hr = i + 8 * half, hc = nt * 16 + rm;
            hb[hr * H_ + hc] = fmaxf(acc1[nt][i], 0.f);
        }
    __syncthreads();

    /* ---- GEMM2: a = relu(h[16x48] @ w2^T[48x768]); out = x + a ---- */
    v2f a2[12];
    #pragma unroll
    for (int ks = 0; ks < 12; ++ks)
        a2[ks] = *(const v2f*)(hb + rm * H_ + ks * 4 + 2 * half);

    for (int nt = 0; nt < 48; ++nt) {
        v8f acc;
        #pragma unroll
        for (int i = 0; i < 8; ++i) acc[i] = 0.f;
        int n = nt * 16 + rm;
        #pragma unroll
        for (int ks = 0; ks < 12; ++ks) {
            v2f bf = *(const v2f*)(w2s + n * H_ + ks * 4 + 2 * half);
            acc = __builtin_amdgcn_wmma_f32_16x16x4_f32(
                false, a2[ks], false, bf, (short)0, acc, false, false);
        }
        #pragma unroll
        for (int i = 0; i < 8; ++i) {
            size_t o = (size_t)(row0 + i + 8 * half) * C_ + nt * 16 + rm;
            out[o] = X[o] + fmaxf(acc[i], 0.f);
        }
    }
}

extern "C" void kernel_launch(void* const* d_in, const int* in_sizes, int n_in,
                              void* d_out, int out_size, void* d_ws, size_t ws_size,
                              hipStream_t stream) {
    const float* X   = (const float*)d_in[0];   // [256,197,768]
    const float* W1  = (const float*)d_in[1];   // [10,48,768]
    const float* W2  = (const float*)d_in[2];   // [10,768,48]
    const float* key = (const float*)d_in[3];   // [10,768]
    float* out = (float*)d_out;
    float* ws  = (float*)d_ws;

    float* meanb = ws;                    // 196608 floats
    float* sim   = ws + 196608;           // 2560 floats
    int*   psidx = (int*)(ws + 199168);   // 256 ints
    int*   major = (int*)(ws + 199424);   // 1 int

    k_mean<<<768, 256, 0, stream>>>(X, meanb);
    k_sim <<<256, 256, 0, stream>>>(meanb, key, sim, psidx);
    k_vote<<<1,   256, 0, stream>>>(sim, psidx, major, out + BIAS_ELEMS);

    size_t sh = (size_t)(2 * H_ * C_ + 8 * 16 * H_) * sizeof(float);  // 319488 B
    k_adapter<<<394, 256, sh, stream>>>(X, W1, W2, major, out);
}